// L_DARK_50251117363510
// MI455X (gfx1250) — compile-verified
//
#include <hip/hip_runtime.h>
#include <math.h>

typedef __attribute__((ext_vector_type(2))) float v2f;
typedef __attribute__((ext_vector_type(8))) float v8f;

#define BATCH 32
#define CHN   3
#define HH    512
#define WW    512
#define KWIN  20
#define PL    9      // left/top pad (K-1)/2 ; right/bottom pad = 10

// vertical tile geometry for kernel 2
#define TROWS 64                  // output rows per tile
#define TCOLS 32                  // columns per tile
#define IROWS (TROWS + KWIN - 1)  // 83 input rows incl. halo
#define NBLK2 (BATCH * (HH / TROWS) * (WW / TCOLS))   // 32*8*16 = 4096

// ---------------------------------------------------------------------------
// Kernel 1: fused channel-min + horizontal 20-wide sliding min (one row/block)
// Log-doubling window build in LDS; edge threads use clamped 20-tap fallback
// (index clamping == replicate padding for a min).
// ---------------------------------------------------------------------------
__global__ void __launch_bounds__(WW)
cmin_hmin_kernel(const float* __restrict__ x, float* __restrict__ tmp) {
  const int bh = blockIdx.x;          // b*512 + h
  const int b  = bh >> 9;
  const int h  = bh & (HH - 1);
  const int w  = threadIdx.x;         // 0..511

  __shared__ float orig[WW];
  __shared__ float sa[WW];
  __shared__ float sb[WW];

  // channel-min first (min commutes with erosion): 96MB -> 32MB working set
  const int off = (b * CHN * HH + h) * WW + w;
  float c0 = x[off];
  float c1 = x[off + HH * WW];
  float c2 = x[off + 2 * HH * WW];
  orig[w] = fminf(fminf(c0, c1), c2);
  __syncthreads();

  // doubling: sa/sb[w] = min over [w .. min(w+len-1, 511)]
  sa[w] = fminf(orig[w], orig[min(w + 1, WW - 1)]);   // len 2
  __syncthreads();
  sb[w] = fminf(sa[w], sa[min(w + 2, WW - 1)]);       // len 4
  __syncthreads();
  sa[w] = fminf(sb[w], sb[min(w + 4, WW - 1)]);       // len 8
  __syncthreads();
  sb[w] = fminf(sa[w], sa[min(w + 8, WW - 1)]);       // len 16
  __syncthreads();

  float r;
  if (w >= PL && w <= WW - 1 - (KWIN - PL - 1)) {
    // interior: [w-9 .. w+10] = [w-9 .. w+6] U [w-5 .. w+10], both unclamped
    r = fminf(sb[w - PL], sb[w - PL + 4]);
  } else {
    r = INFINITY;
    #pragma unroll
    for (int i = 0; i < KWIN; ++i) {
      int idx = min(max(w - PL + i, 0), WW - 1);
      r = fminf(r, orig[idx]);
    }
  }
  tmp[(b * HH + h) * WW + w] = r;
}

// ---------------------------------------------------------------------------
// Kernel 2: tiled vertical 20-tall sliding min via LDS log-doubling
// (read amplification 83/64 = 1.3x instead of 20x), fused with an exact-f32
// WMMA sum reduction: one v_wmma_f32_16x16x4_f32 per wave collapses 64 lane
// partials (D = A x Ones + 0; multiply-by-1.0 is exact).
// ---------------------------------------------------------------------------
__global__ void __launch_bounds__(256)
vmin_wmma_reduce(const float* __restrict__ tmp, float* __restrict__ blockSums) {
  const int idx = blockIdx.x;               // 0..4095
  const int b   = idx >> 7;                 // 128 tiles per image
  const int rt  = (idx >> 4) & 7;           // row tile
  const int ct  = idx & 15;                 // col tile
  const int tb  = rt * TROWS;               // first output row
  const int c0g = ct * TCOLS;               // first column
  const int tid = threadIdx.x;              // 0..255 (8 waves)

  __shared__ float A[IROWS * TCOLS];        // ping
  __shared__ float Bu[IROWS * TCOLS];       // pong
  __shared__ float wsum[8];

  const float* base = tmp + (size_t)b * HH * WW;

  // load 83 halo rows, clamped (== replicate pad); coalesced 128B per wave
  for (int e = tid; e < IROWS * TCOLS; e += 256) {
    int i = e >> 5, c = e & 31;
    int g = min(max(tb - PL + i, 0), HH - 1);
    A[e] = base[g * WW + c0g + c];
  }
  __syncthreads();
  // doubling over rows: buf[i] = min over tile rows [i .. i+len-1] (exact, no
  // clamps needed: ranges chosen so i+len-1 <= 82)
  for (int e = tid; e < 82 * TCOLS; e += 256) {          // len 2, i<=81
    int i = e >> 5, c = e & 31;
    Bu[e] = fminf(A[i * 32 + c], A[(i + 1) * 32 + c]);
  }
  __syncthreads();
  for (int e = tid; e < 80 * TCOLS; e += 256) {          // len 4, i<=79
    int i = e >> 5, c = e & 31;
    A[e] = fminf(Bu[i * 32 + c], Bu[(i + 2) * 32 + c]);
  }
  __syncthreads();
  for (int e = tid; e < 76 * TCOLS; e += 256) {          // len 8, i<=75
    int i = e >> 5, c = e & 31;
    Bu[e] = fminf(A[i * 32 + c], A[(i + 4) * 32 + c]);
  }
  __syncthreads();
  for (int e = tid; e < 68 * TCOLS; e += 256) {          // len 16, i<=67
    int i = e >> 5, c = e & 31;
    A[e] = fminf(Bu[i * 32 + c], Bu[(i + 8) * 32 + c]);
  }
  __syncthreads();

  // outputs: out[r] = min(m16[r], m16[r+4]) covers tile rows [r..r+19]
  // = global rows [h-9..h+10] clamped. 8 outputs/thread, fixed-order partials.
  float p0 = 0.0f, p1 = 0.0f;
  #pragma unroll
  for (int k = 0; k < 8; ++k) {
    int e = tid + k * 256;                  // exactly 2048 = 64*32 outputs
    int i = e >> 5, c = e & 31;
    float v = fminf(A[i * 32 + c], A[(i + 4) * 32 + c]);
    if (k < 4) p0 += v; else p1 += v;
  }

  float waveTotal;
#if __has_builtin(__builtin_amdgcn_wmma_f32_16x16x4_f32)
  // A (16x4 f32) = 2 values/lane; B = ones; EXEC all-1s (no divergence here)
  v2f a;    a[0] = p0;   a[1] = p1;
  v2f ones; ones[0] = 1.0f; ones[1] = 1.0f;
  v8f c8 = {};
  c8 = __builtin_amdgcn_wmma_f32_16x16x4_f32(
        /*neg_a=*/false, a, /*neg_b=*/false, ones,
        /*c_mod=*/(short)0, c8, /*reuse_a=*/false, /*reuse_b=*/false);
  // columns of D are identical (B==1): lanes 0-15 hold rows 0-7 sum, lanes
  // 16-31 rows 8-15; the two halves together give the full wave sum.
  float s = c8[0] + c8[1] + c8[2] + c8[3] + c8[4] + c8[5] + c8[6] + c8[7];
  waveTotal = s + __shfl_xor(s, 16, 32);
#else
  float s = p0 + p1;
  #pragma unroll
  for (int m = 16; m >= 1; m >>= 1) s += __shfl_xor(s, m, 32);
  waveTotal = s;
#endif

  const int lane = tid & 31;
  const int wave = tid >> 5;                // wave32: 8 waves / 256 threads
  if (lane == 0) wsum[wave] = waveTotal;
  __syncthreads();
  if (tid == 0) {
    float bs = 0.0f;
    #pragma unroll
    for (int i = 0; i < 8; ++i) bs += wsum[i];
    blockSums[idx] = bs;                    // deterministic: no float atomics
  }
}

// ---------------------------------------------------------------------------
// Kernel 3: fixed-order reduction of 4096 block sums -> |1 - mean|
// ---------------------------------------------------------------------------
__global__ void __launch_bounds__(256)
finalize_kernel(const float* __restrict__ blockSums, float* __restrict__ out) {
  __shared__ float red[256];
  const int t = threadIdx.x;
  float s = 0.0f;
  for (int i = t; i < NBLK2; i += 256) s += blockSums[i];
  red[t] = s;
  __syncthreads();
  #pragma unroll
  for (int offd = 128; offd > 0; offd >>= 1) {
    if (t < offd) red[t] += red[t + offd];
    __syncthreads();
  }
  if (t == 0) {
    const float invN = 1.0f / 8388608.0f;   // 1/(32*512*512), exact pow2
    out[0] = fabsf(1.0f - red[0] * invN);
  }
}

extern "C" void kernel_launch(void* const* d_in, const int* in_sizes, int n_in,
                              void* d_out, int out_size, void* d_ws, size_t ws_size,
                              hipStream_t stream) {
  const float* x = (const float*)d_in[0];       // [32,3,512,512] f32
  float* out = (float*)d_out;                   // 1 f32 scalar

  // workspace layout: [0, 64KB) block sums; [64KB, 64KB+32MB) intermediate
  float* blockSums = (float*)d_ws;
  float* tmp = (float*)((char*)d_ws + (64 << 10));

  cmin_hmin_kernel<<<BATCH * HH, WW, 0, stream>>>(x, tmp);
  vmin_wmma_reduce<<<NBLK2, 256, 0, stream>>>(tmp, blockSums);
  finalize_kernel<<<1, 256, 0, stream>>>(blockSums, out);
}